// MODEL_BH_43181601194155
// MI455X (gfx1250) — compile-verified
//
#include <hip/hip_runtime.h>
#include <math.h>

// DKVMN forward for MI455X (gfx1250, wave32).
// f32 WMMA (V_WMMA_F32_16X16X4_F32) used for all GEMMs: exact f32 math,
// 16x16 C tile per wave, K stepped by 4 (A/B fragments = float2 per lane).

typedef __attribute__((ext_vector_type(2))) float v2f;
typedef __attribute__((ext_vector_type(8))) float v8f;

#define Bsz 256
#define Ssz 500
#define DQ  64
#define DVd 128
#define Mm  64
#define FCn 64
#define BT  16      // batch tile per workgroup (WMMA M)
#define DVS 16      // DV slice per workgroup
#define BS  (Bsz*Ssz)
#define PRED_ROWS 16

// ---- scan-kernel LDS layout (float offsets, dynamic LDS) ----
#define O_MK   0                      // 64*64   memory key
#define O_EW   (O_MK + Mm*DQ)         // 128*16  erase_W slice
#define O_AW   (O_EW + DVd*DVS)       // 128*16  add_W slice
#define O_MV   (O_AW + DVd*DVS)       // 16*64*16 recurrent Mv state
#define O_Q    (O_MV + BT*Mm*DVS)     // 16*64   q_e tile
#define O_QA   (O_Q  + BT*DQ)         // 16*128  qa_e tile
#define O_W    (O_QA + BT*DVd)        // 16*64   attention weights
#define O_E    (O_W  + BT*Mm)         // 16*16   erase gate
#define O_A    (O_E  + BT*DVS)        // 16*16   add vector
#define O_QI   (O_A  + BT*DVS)        // 16 ints
#define O_QAI  (O_QI + BT)            // 16 ints
#define O_MASK (O_QAI+ BT)            // 16
#define O_EB   (O_MASK+BT)            // 16
#define O_AB   (O_EB + DVS)           // 16
#define SCAN_LDS_FLOATS (O_AB + DVS)  // = 29264 floats = ~114.3 KB (<320KB/WGP)

static __device__ __forceinline__ v8f wmma_f32_k4(v8f c, v2f a, v2f b) {
  return __builtin_amdgcn_wmma_f32_16x16x4_f32(false, a, false, b, (short)0, c,
                                               false, false);
}

// One workgroup owns 16 batch rows x a 16-wide DV slice; Mv state lives in LDS
// for all 500 sequential steps. 4 waves of 32.
__global__ __launch_bounds__(128) void dkvmn_scan(
    const int* __restrict__ q_data, const int* __restrict__ qa_data,
    const float* __restrict__ q_embed_w, const float* __restrict__ qa_embed_w,
    const float* __restrict__ Mk, const float* __restrict__ Mv0,
    const float* __restrict__ erase_W, const float* __restrict__ erase_b,
    const float* __restrict__ add_W, const float* __restrict__ add_b,
    float* __restrict__ reads_ws) {
  extern __shared__ float sm[];
  const int tid  = threadIdx.x;
  const int wave = tid >> 5;
  const int lane = tid & 31;
  const int b0 = blockIdx.x * BT;   // batch tile base
  const int d0 = blockIdx.y * DVS;  // DV slice base

  float* sMk = sm + O_MK;
  float* sEW = sm + O_EW;
  float* sAW = sm + O_AW;
  float* sMv = sm + O_MV;
  float* sQ  = sm + O_Q;
  float* sQA = sm + O_QA;
  float* sW  = sm + O_W;
  float* sE  = sm + O_E;
  float* sA  = sm + O_A;
  int*   sQi = (int*)(sm + O_QI);
  int*   sQAi= (int*)(sm + O_QAI);
  float* sMask = sm + O_MASK;
  float* sEB = sm + O_EB;
  float* sAB = sm + O_AB;

  // ---- one-time loads ----
  for (int e = tid; e < Mm * DQ; e += 128) sMk[e] = Mk[e];
  for (int e = tid; e < DVd * DVS; e += 128) {
    int k = e / DVS, j = e % DVS;
    sEW[e] = erase_W[k * DVd + d0 + j];
    sAW[e] = add_W[k * DVd + d0 + j];
  }
  for (int e = tid; e < BT * Mm * DVS; e += 128) {
    int d = e % DVS;
    int m = (e / DVS) % Mm;
    sMv[e] = Mv0[m * DVd + d0 + d];  // broadcast init state to every batch row
  }
  if (tid < DVS) { sEB[tid] = erase_b[d0 + tid]; sAB[tid] = add_b[d0 + tid]; }
  __syncthreads();

  // WMMA fragment decoding (per ISA layout): A 16x4 f32, B 4x16 f32
  const int arow  = lane & 15;            // A matrix row
  const int koff  = (lane >> 4) * 2;      // K pair: lanes16-31 hold K=2,3
  const int rbase = (lane >> 4) * 8;      // C rows: lanes16-31 hold M=8..15
  const int ncol  = lane & 15;            // B/C column within tile

  for (int t = 0; t < Ssz; ++t) {
    // ---- fetch indices + embedding tiles for this step ----
    if (tid < BT) {
      int qi  = q_data[(b0 + tid) * Ssz + t];
      int qai = qa_data[(b0 + tid) * Ssz + t];
      sQi[tid] = qi; sQAi[tid] = qai;
      sMask[tid] = (qi >= 1) ? 1.0f : 0.0f;
    }
    __syncthreads();
    for (int e = tid; e < BT * DQ; e += 128)
      sQ[e] = q_embed_w[sQi[e >> 6] * DQ + (e & 63)];
    for (int e = tid; e < BT * DVd; e += 128)
      sQA[e] = qa_embed_w[sQAi[e >> 7] * DVd + (e & 127)];
    __syncthreads();

    // ---- w logits: (16 x 64) = qTile @ Mk^T ; wave n owns n-tile ----
    {
      v8f c = {0.f, 0.f, 0.f, 0.f, 0.f, 0.f, 0.f, 0.f};
      const int n0 = wave * 16 + ncol;  // global column = memory slot m
#pragma unroll
      for (int kk = 0; kk < DQ / 4; ++kk) {
        v2f a  = {sQ[arow * DQ + kk * 4 + koff], sQ[arow * DQ + kk * 4 + koff + 1]};
        v2f bm = {sMk[n0 * DQ + kk * 4 + koff], sMk[n0 * DQ + kk * 4 + koff + 1]};
        c = wmma_f32_k4(c, a, bm);
      }
#pragma unroll
      for (int r = 0; r < 8; ++r)
        sW[(r + rbase) * Mm + wave * 16 + ncol] = c[r];
    }
    __syncthreads();

    // ---- softmax over the 64 memory slots (one thread per batch row) ----
    if (tid < BT) {
      float mx = -1e30f;
      for (int m = 0; m < Mm; ++m) mx = fmaxf(mx, sW[tid * Mm + m]);
      float sum = 0.f;
      for (int m = 0; m < Mm; ++m) {
        float v = expf(sW[tid * Mm + m] - mx);
        sW[tid * Mm + m] = v; sum += v;
      }
      float inv = 1.0f / sum;
      for (int m = 0; m < Mm; ++m) sW[tid * Mm + m] *= inv;
    }
    __syncthreads();

    // ---- erase/add gates: (16 x 16) = qaTile @ W_slice ; wave0=e wave1=a ----
    if (wave < 2) {
      const float* Bm = wave ? sAW : sEW;
      v8f c = {0.f, 0.f, 0.f, 0.f, 0.f, 0.f, 0.f, 0.f};
#pragma unroll
      for (int kk = 0; kk < DVd / 4; ++kk) {
        v2f a  = {sQA[arow * DVd + kk * 4 + koff], sQA[arow * DVd + kk * 4 + koff + 1]};
        v2f bm = {Bm[(kk * 4 + koff) * DVS + ncol], Bm[(kk * 4 + koff + 1) * DVS + ncol]};
        c = wmma_f32_k4(c, a, bm);
      }
      float* dst = wave ? sA : sE;
      const float* bias = wave ? sAB : sEB;
#pragma unroll
      for (int r = 0; r < 8; ++r) {
        float x = c[r] + bias[ncol];
        dst[(r + rbase) * DVS + ncol] =
            wave ? tanhf(x) : (1.0f / (1.0f + expf(-x)));
      }
    }
    __syncthreads();

    // ---- fused read (w . Mv, old state) + masked Mv update ----
    for (int p = tid; p < BT * DVS; p += 128) {
      int b = p >> 4, d = p & 15;
      float mval = sMask[b];
      float ev = sE[b * DVS + d], av = sA[b * DVS + d];
      float acc = 0.f;
      for (int m = 0; m < Mm; ++m) {
        float wv = sW[b * Mm + m];
        float mv = sMv[(b * Mm + m) * DVS + d];
        acc += wv * mv;                                  // read uses OLD Mv
        float nv = mv * (1.0f - wv * ev) + wv * av;
        sMv[(b * Mm + m) * DVS + d] = mval * nv + (1.0f - mval) * mv;
      }
      reads_ws[((size_t)(b0 + b) * Ssz + t) * DVd + d0 + d] = acc;
    }
    __syncthreads();
  }
}

// Prediction MLP: 16 rows x 64 cols per workgroup, K=192 via WMMA + bh1 rank-1
// epilogue; deterministic per-block partial BCE sums (no float atomics).
__global__ __launch_bounds__(128) void dkvmn_predict(
    const int* __restrict__ q_data, const float* __restrict__ q_embed_w,
    const float* __restrict__ bh1, const float* __restrict__ target,
    const float* __restrict__ read_W, const float* __restrict__ read_b,
    const float* __restrict__ pred_W, const float* __restrict__ pred_b,
    const float* __restrict__ reads_ws, float* __restrict__ d_out,
    float* __restrict__ partials) {
  __shared__ float sIn[PRED_ROWS * 197];  // 193-wide input, odd stride
  __shared__ float sH[PRED_ROWS * FCn];
  __shared__ float sRB[FCn];
  __shared__ float sPW[FCn];
  __shared__ int   sQi[PRED_ROWS];
  __shared__ float sBce[PRED_ROWS];
  __shared__ float sMsk[PRED_ROWS];

  const int tid = threadIdx.x, wave = tid >> 5, lane = tid & 31;
  const int row0 = blockIdx.x * PRED_ROWS;

  if (tid < FCn) { sRB[tid] = read_b[tid]; sPW[tid] = pred_W[tid]; }
  if (tid < PRED_ROWS) sQi[tid] = q_data[row0 + tid];
  __syncthreads();
  for (int e = tid; e < PRED_ROWS * DVd; e += 128)
    sIn[(e >> 7) * 197 + (e & 127)] =
        reads_ws[(size_t)(row0 + (e >> 7)) * DVd + (e & 127)];
  for (int e = tid; e < PRED_ROWS * DQ; e += 128)
    sIn[(e >> 6) * 197 + 128 + (e & 63)] = q_embed_w[sQi[e >> 6] * DQ + (e & 63)];
  if (tid < PRED_ROWS) sIn[tid * 197 + 192] = bh1[row0 + tid];
  __syncthreads();

  const int arow = lane & 15, koff = (lane >> 4) * 2, rbase = (lane >> 4) * 8;
  const int n0 = wave * 16 + (lane & 15);
  v8f c = {0.f, 0.f, 0.f, 0.f, 0.f, 0.f, 0.f, 0.f};
#pragma unroll
  for (int kk = 0; kk < 48; ++kk) {  // K = 192 (B fragments from L2-cached W)
    v2f a  = {sIn[arow * 197 + kk * 4 + koff], sIn[arow * 197 + kk * 4 + koff + 1]};
    v2f bm = {read_W[(kk * 4 + koff) * FCn + n0],
              read_W[(kk * 4 + koff + 1) * FCn + n0]};
    c = wmma_f32_k4(c, a, bm);
  }
  const float w192 = read_W[192 * FCn + n0];
#pragma unroll
  for (int r = 0; r < 8; ++r) {
    int row = r + rbase;
    float x = c[r] + sIn[row * 197 + 192] * w192 + sRB[n0];
    sH[row * FCn + n0] = tanhf(x);
  }
  __syncthreads();
  if (tid < PRED_ROWS) {
    float acc = pred_b[0];
    for (int k = 0; k < FCn; ++k) acc += sH[tid * FCn + k] * sPW[k];
    int row = row0 + tid;
    float tv = target[row];
    d_out[1 + row] = 1.0f / (1.0f + expf(-acc));  // sigmoid(logits)
    d_out[1 + BS + row] = tv;                      // t passthrough
    float mk = (tv >= 0.0f) ? 1.0f : 0.0f;
    float bce = fmaxf(acc, 0.0f) - acc * tv + log1pf(expf(-fabsf(acc)));
    sBce[tid] = bce * mk; sMsk[tid] = mk;
  }
  __syncthreads();
  if (tid == 0) {  // fixed-order in-block reduction -> deterministic
    float n = 0.f, d = 0.f;
    for (int i = 0; i < PRED_ROWS; ++i) { n += sBce[i]; d += sMsk[i]; }
    partials[blockIdx.x * 2 + 0] = n;
    partials[blockIdx.x * 2 + 1] = d;
  }
}

__global__ void dkvmn_loss(const float* __restrict__ partials,
                           float* __restrict__ d_out) {
  if (threadIdx.x == 0 && blockIdx.x == 0) {
    float n = 0.f, d = 0.f;
    for (int i = 0; i < BS / PRED_ROWS; ++i) {
      n += partials[2 * i]; d += partials[2 * i + 1];
    }
    d_out[0] = n / fmaxf(d, 1.0f);
  }
}

extern "C" void kernel_launch(void* const* d_in, const int* in_sizes, int n_in,
                              void* d_out, int out_size, void* d_ws,
                              size_t ws_size, hipStream_t stream) {
  (void)in_sizes; (void)n_in; (void)out_size; (void)ws_size;
  const int*   q_data     = (const int*)d_in[0];
  const int*   qa_data    = (const int*)d_in[1];
  const float* target     = (const float*)d_in[2];
  const float* bh1        = (const float*)d_in[3];
  // d_in[4..6] (bh2..bh4) unused by the reference computation
  const float* q_embed_w  = (const float*)d_in[7];
  const float* qa_embed_w = (const float*)d_in[8];
  const float* Mk         = (const float*)d_in[9];
  const float* Mv0        = (const float*)d_in[10];
  const float* erase_W    = (const float*)d_in[11];
  const float* erase_b    = (const float*)d_in[12];
  const float* add_W      = (const float*)d_in[13];
  const float* add_b      = (const float*)d_in[14];
  const float* read_W     = (const float*)d_in[15];
  const float* read_b     = (const float*)d_in[16];
  const float* pred_W     = (const float*)d_in[17];
  const float* pred_b     = (const float*)d_in[18];

  float* ws       = (float*)d_ws;
  float* reads_ws = ws;                        // B*S*DV f32 = 62.5 MB
  float* partials = ws + (size_t)BS * DVd;     // 2 * 8000 f32

  dkvmn_scan<<<dim3(Bsz / BT, DVd / DVS), 128,
               SCAN_LDS_FLOATS * sizeof(float), stream>>>(
      q_data, qa_data, q_embed_w, qa_embed_w, Mk, Mv0, erase_W, erase_b, add_W,
      add_b, reads_ws);
  dkvmn_predict<<<BS / PRED_ROWS, 128, 0, stream>>>(
      q_data, q_embed_w, bh1, target, read_W, read_b, pred_W, pred_b, reads_ws,
      (float*)d_out, partials);
  dkvmn_loss<<<1, 32, 0, stream>>>(partials, (float*)d_out);
}